// cheb_conv_with_Att_GL_687194768238
// MI455X (gfx1250) — compile-verified
//
#include <hip/hip_runtime.h>
#include <hip/hip_bf16.h>

// cheb_conv_with_Att_GL for MI455X (gfx1250, wave32, WMMA f32 16x16x4)
// B=16, T=16, V=512, F=64, O=64, K_ORDER=3

typedef __attribute__((ext_vector_type(2))) float v2f;
typedef __attribute__((ext_vector_type(8))) float v8f;

#define BT 256
#define V 512
#define F 64
#define O 64

static __device__ __forceinline__ v8f wmma_f32(v2f a, v2f b, v8f c) {
  return __builtin_amdgcn_wmma_f32_16x16x4_f32(false, a, false, b, (short)0, c,
                                               false, false);
}

// ---------------------------------------------------------------------------
// Kernel A: G[k][bt][u][o] = sum_f x[bt][u][f] * Theta[k][f][o], k = 0..2
// ---------------------------------------------------------------------------
__global__ __launch_bounds__(256) void gmat_kernel(const float* __restrict__ x,
                                                   const float* __restrict__ Theta,
                                                   float* __restrict__ G) {
  const int bt = blockIdx.x;
  const int tid = threadIdx.x;
  const int lane = tid & 31;
  const int wave = tid >> 5;
  const int g = lane >> 4;   // lane group: 0 (lanes 0-15), 1 (lanes 16-31)
  const int nl = lane & 15;

  __shared__ float th[3][64][65];  // Theta staged, padded
  __shared__ float xs[64][65];     // x chunk staged, padded

  for (int idx = tid; idx < 3 * 64 * 64; idx += 256) {
    int k = idx >> 12;
    int rem = idx & 4095;
    th[k][rem >> 6][rem & 63] = Theta[idx];
  }

  const float* xbt = x + (size_t)bt * V * F;

  for (int uc = 0; uc < V; uc += 64) {
    __syncthreads();
    for (int idx = tid; idx < 64 * 64; idx += 256) {
      int r = idx >> 6;
      int o = idx & 63;
      xs[r][o] = xbt[(size_t)(uc + r) * F + o];
    }
    __syncthreads();

    // 48 jobs: (ut 0..3) x (k 0..2) x (ot 0..3); wave w takes jobs w+8*jj
    for (int jj = 0; jj < 6; ++jj) {
      int job = jj * 8 + wave;
      int ut = job / 12;
      int rem = job % 12;
      int k = rem >> 2;
      int ot = rem & 3;

      v8f acc = {0.f, 0.f, 0.f, 0.f, 0.f, 0.f, 0.f, 0.f};
      for (int fk = 0; fk < 64; fk += 4) {
        v2f a, b;
        // A[M=u-row, K=f]: lane nl = row, K = fk + 2*g + j
        a.x = xs[ut * 16 + nl][fk + 2 * g + 0];
        a.y = xs[ut * 16 + nl][fk + 2 * g + 1];
        // B[K=f, N=o]
        b.x = th[k][fk + 2 * g + 0][ot * 16 + nl];
        b.y = th[k][fk + 2 * g + 1][ot * 16 + nl];
        acc = wmma_f32(a, b, acc);
      }
      float* Gd = G + ((size_t)k * BT + bt) * V * O + (size_t)(uc + ut * 16) * O +
                  ot * 16;
#pragma unroll
      for (int r = 0; r < 8; ++r) {
        // D layout: reg r -> M = r + 8*g, N = nl
        Gd[(size_t)(r + 8 * g) * O + nl] = acc[r];
      }
    }
  }
}

// ---------------------------------------------------------------------------
// Kernel B: deg[bt][v] = sum_u min(S[bt][u][v], S[bt][v][u])
// Triangle-tiled: each S element read exactly once.
// ---------------------------------------------------------------------------
__global__ __launch_bounds__(256) void deg_kernel(const float* __restrict__ S,
                                                  float* __restrict__ deg) {
  const int bt = blockIdx.x;
  const int tid = threadIdx.x;
  __shared__ float a[64][65];
  __shared__ float bsh[64][65];
  __shared__ float dl[V];

  for (int i = tid; i < V; i += 256) dl[i] = 0.f;

  const float* Sbt = S + (size_t)bt * V * V;
  const int c = tid & 63;
  const int r0 = tid >> 6;  // 0..3

  for (int I = 0; I < 8; ++I) {
    for (int J = I; J < 8; ++J) {
      __syncthreads();
      for (int k = 0; k < 16; ++k) {
        int r = k * 4 + r0;
        a[r][c] = Sbt[(size_t)(I * 64 + r) * V + J * 64 + c];
        if (J > I) bsh[r][c] = Sbt[(size_t)(J * 64 + r) * V + I * 64 + c];
      }
      __syncthreads();
      if (I == J) {
        float p = 0.f;
        for (int k = 0; k < 16; ++k) {
          int r = k * 4 + r0;
          p += fminf(a[r][c], a[c][r]);
        }
        atomicAdd(&dl[J * 64 + c], p);
      } else {
        float p1 = 0.f, p2 = 0.f;
        for (int k = 0; k < 16; ++k) {
          int r = k * 4 + r0;
          p1 += fminf(a[r][c], bsh[c][r]);   // tile (I,J): cols in J block
          p2 += fminf(bsh[r][c], a[c][r]);   // tile (J,I): cols in I block
        }
        atomicAdd(&dl[J * 64 + c], p1);
        atomicAdd(&dl[I * 64 + c], p2);
      }
    }
  }
  __syncthreads();
  for (int i = tid; i < V; i += 256) deg[(size_t)bt * V + i] = dl[i];
}

// ---------------------------------------------------------------------------
// Kernel C: out[bt][v][o] = relu( sum_u W1[u,v]*G1[u,o] + W2[u,v]*G2[u,o]
//                                 + Att[v,v]*G0[v,o] )
// W1 = L .* Att, W2 = (2*L.*L - I) .* Att, L = diag(deg-1) - min(S,S^T)
// Double-buffered LDS staging: next chunk's G1/G2/Sv loads issued into
// registers before the current chunk's WMMA loop; one barrier per chunk.
// ---------------------------------------------------------------------------
__global__ __launch_bounds__(256) void main_kernel(const float* __restrict__ S,
                                                   const float* __restrict__ Att,
                                                   const float* __restrict__ G,
                                                   const float* __restrict__ deg,
                                                   float* __restrict__ out) {
  const int bt = blockIdx.x;
  const int b = bt >> 4;
  const int tid = threadIdx.x;
  const int lane = tid & 31;
  const int wave = tid >> 5;
  const int g = lane >> 4;
  const int nl = lane & 15;

  __shared__ float g1[2][64][65];
  __shared__ float g2[2][64][65];
  __shared__ float svs[2][8][16][65];  // per-wave transposed S strip
  __shared__ float dl[V];

  const float* Sbt = S + (size_t)bt * V * V;
  const float* Ab = Att + (size_t)b * V * V;
  const float* G0 = G + (size_t)bt * V * O;
  const float* G1 = G + (size_t)(BT + bt) * V * O;
  const float* G2 = G + (size_t)(2 * BT + bt) * V * O;

  for (int i = tid; i < V; i += 256) dl[i] = deg[(size_t)bt * V + i];

  // staging index decomposition (constant per thread)
  const int gro = tid >> 6;  // G-chunk row base (0..3), rows gro + 4*q
  const int go = tid & 63;   // G-chunk column

  for (int i = 0; i < 4; ++i) {
    const int v0 = (i * 8 + wave) * 16;  // this wave's 16 output rows (v)
    v8f acc[4];
#pragma unroll
    for (int ot = 0; ot < 4; ++ot)
      acc[ot] = (v8f){0.f, 0.f, 0.f, 0.f, 0.f, 0.f, 0.f, 0.f};

    float rg1[16], rg2[16], rsv[32];

    // ---- prologue: load chunk 0 into regs, store to buffer 0 ----
#pragma unroll
    for (int q = 0; q < 16; ++q) {
      int r = gro + 4 * q;
      rg1[q] = G1[(size_t)r * O + go];
      rg2[q] = G2[(size_t)r * O + go];
    }
#pragma unroll
    for (int q = 0; q < 32; ++q) {
      int r = q >> 1;
      int cc = lane + 32 * (q & 1);
      rsv[q] = Sbt[(size_t)(v0 + r) * V + cc];
    }
#pragma unroll
    for (int q = 0; q < 16; ++q) {
      int r = gro + 4 * q;
      g1[0][r][go] = rg1[q];
      g2[0][r][go] = rg2[q];
    }
#pragma unroll
    for (int q = 0; q < 32; ++q) {
      int r = q >> 1;
      int cc = lane + 32 * (q & 1);
      svs[0][wave][r][cc] = rsv[q];
    }
    __syncthreads();

    for (int c = 0; c < 8; ++c) {
      const int cur = c & 1;
      const int nxt = cur ^ 1;
      const int uc = c * 64;

      // ---- issue next chunk's global loads (overlap with compute) ----
      if (c < 7) {
        const int un = uc + 64;
#pragma unroll
        for (int q = 0; q < 16; ++q) {
          int r = un + gro + 4 * q;
          rg1[q] = G1[(size_t)r * O + go];
          rg2[q] = G2[(size_t)r * O + go];
        }
#pragma unroll
        for (int q = 0; q < 32; ++q) {
          int r = q >> 1;
          int cc = un + lane + 32 * (q & 1);
          rsv[q] = Sbt[(size_t)(v0 + r) * V + cc];
        }
      }

      // ---- compute on current buffer ----
      for (int ku = 0; ku < 64; ku += 4) {
        v2f a1, a2;
        const int v = v0 + nl;
#pragma unroll
        for (int j = 0; j < 2; ++j) {
          int uu = uc + ku + 2 * g + j;  // A layout: K = j + 2*laneGroup
          float suv = Sbt[(size_t)uu * V + v];              // coalesced row read
          float svu = svs[cur][wave][nl][ku + 2 * g + j];   // LDS, conflict-free
          float m = fminf(suv, svu);
          float att = Ab[(size_t)uu * V + v];
          float diag = (uu == v) ? 1.0f : 0.0f;
          float lval = diag * (dl[uu] - 1.0f) - m;
          float w1 = lval * att;
          float w2 = (2.0f * lval * lval - diag) * att;
          if (j == 0) {
            a1.x = w1;
            a2.x = w2;
          } else {
            a1.y = w1;
            a2.y = w2;
          }
        }
#pragma unroll
        for (int ot = 0; ot < 4; ++ot) {
          v2f b1, b2;
          b1.x = g1[cur][ku + 2 * g + 0][ot * 16 + nl];
          b1.y = g1[cur][ku + 2 * g + 1][ot * 16 + nl];
          b2.x = g2[cur][ku + 2 * g + 0][ot * 16 + nl];
          b2.y = g2[cur][ku + 2 * g + 1][ot * 16 + nl];
          acc[ot] = wmma_f32(a1, b1, acc[ot]);
          acc[ot] = wmma_f32(a2, b2, acc[ot]);
        }
      }

      // ---- store next chunk regs into alternate buffer; one barrier ----
      if (c < 7) {
#pragma unroll
        for (int q = 0; q < 16; ++q) {
          int r = gro + 4 * q;
          g1[nxt][r][go] = rg1[q];
          g2[nxt][r][go] = rg2[q];
        }
#pragma unroll
        for (int q = 0; q < 32; ++q) {
          int r = q >> 1;
          int cc = lane + 32 * (q & 1);
          svs[nxt][wave][r][cc] = rsv[q];
        }
      }
      __syncthreads();
    }

    // ---- epilogue: diagonal k=0 term + ReLU + store ----
#pragma unroll
    for (int ot = 0; ot < 4; ++ot) {
#pragma unroll
      for (int r = 0; r < 8; ++r) {
        int v = v0 + r + 8 * g;  // D layout: M = r + 8*laneGroup
        int o = ot * 16 + nl;
        float g0 = G0[(size_t)v * O + o];
        float avv = Ab[(size_t)v * V + v];
        float val = acc[ot][r] + avv * g0;
        out[(size_t)bt * V * O + (size_t)v * O + o] = fmaxf(val, 0.f);
      }
    }
  }
}

// ---------------------------------------------------------------------------
extern "C" void kernel_launch(void* const* d_in, const int* in_sizes, int n_in,
                              void* d_out, int out_size, void* d_ws, size_t ws_size,
                              hipStream_t stream) {
  const float* x = (const float*)d_in[0];      // (16,16,512,64)
  const float* Att = (const float*)d_in[1];    // (16,512,512)
  const float* S = (const float*)d_in[2];      // (16,16,512,512)
  const float* Theta = (const float*)d_in[3];  // (3,64,64)

  float* G = (float*)d_ws;                       // 3 * 256 * 512 * 64 floats
  float* deg = G + (size_t)3 * BT * V * O;       // 256 * 512 floats

  gmat_kernel<<<BT, 256, 0, stream>>>(x, Theta, G);
  deg_kernel<<<BT, 256, 0, stream>>>(S, deg);
  main_kernel<<<BT, 256, 0, stream>>>(S, Att, G, deg, (float*)d_out);
}